// AttentionLayer_56590489092321
// MI455X (gfx1250) — compile-verified
//
#include <hip/hip_runtime.h>
#include <hip/hip_bf16.h>
#include <math.h>

// ---------------------------------------------------------------------------
// AttentionLayer (Bahdanau-style scaled dot attention) for MI455X / gfx1250.
//
// Refactor: energy[b,l] = (enc[b,l,:] . qk[b,:] + cb[b]) / sqrt(D)
//   where qk = (h@Wq + bq) @ Wk^T, cb[b] = (h@Wq+bq)[b,:] . bk
// => encoder_output (256 MB) is streamed exactly ONCE with an online-softmax
//    + context accumulation (flash style). HBM-roofline ~11 us @ 23.3 TB/s.
// Projections (32x1024x1024 GEMMs) use V_WMMA_F32_16X16X4_F32 (exact fp32).
// ---------------------------------------------------------------------------

typedef __attribute__((ext_vector_type(2))) float v2f;
typedef __attribute__((ext_vector_type(4))) float v4f;
typedef __attribute__((ext_vector_type(8))) float v8f;

#define SPLIT 32         // L-splits per batch row for the streaming pass
#define PART_STRIDE 1032 // 1024 acc + m + s, padded, 16B-aligned stride

// ---------------------------------------------------------------------------
// fp32 WMMA GEMM: C[M,N] = A[M,K] @ B'(K,N) + bias, B' = (OPB ? B^T : B).
// OPB is a template parameter so the inner K loop is branch-free:
// 2x global_load_b64 + v_wmma_f32_16x16x4_f32 per K-step of 4, unrolled x2.
// One wave (32 lanes) computes one 16x16 tile.
// ISA 7.12.2 f32 layouts:
//   A 16x4 : lanes 0-15 -> M=lane, K={k0,k0+1}; lanes 16-31 -> K={k0+2,k0+3}
//   B 4x16 : symmetric (lane&15 = N, half selects K pair)
//   C 16x16: VGPR r -> (M = r + 8*(lane>>4), N = lane&15)
// ---------------------------------------------------------------------------
template <int OPB>
__global__ __launch_bounds__(32)
void attn_wmma_gemm_f32(const float* __restrict__ A,
                        const float* __restrict__ Bm,
                        const float* __restrict__ bias,
                        float* __restrict__ C,
                        int M, int N, int K) {
  const int tn   = blockIdx.x;        // N/16 tiles
  const int tm   = blockIdx.y;        // M/16 tiles
  const int lane = threadIdx.x;       // 0..31, one wave per block
  const int half = lane >> 4;         // 0 or 1
  const int l15  = lane & 15;
  const int m = tm * 16 + l15;        // A row for this lane
  const int n = tn * 16 + l15;        // B col for this lane

  const float* __restrict__ aRow = A + (size_t)m * K;

  v8f acc = {};
  for (int k0 = 0; k0 < K; k0 += 8) {   // two WMMA K-steps per iteration
#pragma unroll
    for (int u = 0; u < 2; ++u) {
      const int ka = k0 + u * 4 + half * 2;
      v2f a, b;
      a.x = aRow[ka];
      a.y = aRow[ka + 1];
      if (OPB) {                        // B stored [N,K]; B'[k][n] = B[n*K+k]
        b.x = Bm[(size_t)n * K + ka];
        b.y = Bm[(size_t)n * K + ka + 1];
      } else {                          // B stored [K,N]
        b.x = Bm[(size_t)ka * N + n];
        b.y = Bm[(size_t)(ka + 1) * N + n];
      }
      acc = __builtin_amdgcn_wmma_f32_16x16x4_f32(
          /*neg_a=*/false, a, /*neg_b=*/false, b,
          /*c_mod=*/(short)0, acc, /*reuse_a=*/false, /*reuse_b=*/false);
    }
  }

  const float bv = bias ? bias[n] : 0.0f;
  const int mbase = tm * 16 + half * 8;
#pragma unroll
  for (int r = 0; r < 8; ++r) {
    C[(size_t)(mbase + r) * N + n] = acc[r] + bv;
  }
}

// cb[b] = q[b,:] . bk   (tiny)
__global__ void attn_row_dot(const float* __restrict__ q,
                             const float* __restrict__ bk,
                             float* __restrict__ cb, int Dn) {
  const int b = blockIdx.x;
  float p = 0.0f;
  for (int d = threadIdx.x; d < Dn; d += blockDim.x)
    p += q[(size_t)b * Dn + d] * bk[d];
  __shared__ float red[256];
  red[threadIdx.x] = p;
  __syncthreads();
  for (int s = 128; s > 0; s >>= 1) {
    if (threadIdx.x < s) red[threadIdx.x] += red[threadIdx.x + s];
    __syncthreads();
  }
  if (threadIdx.x == 0) cb[b] = red[0];
}

// NT 128-bit load: enc is a one-shot 256MB stream; keep it out of L2's way.
// __builtin_nontemporal_load requires a clang ext-vector, not HIP float4.
static __device__ __forceinline__ float4 ntload4(const float4* p) {
  v4f t = __builtin_nontemporal_load((const v4f*)p);
  return make_float4(t.x, t.y, t.z, t.w);
}

// ---------------------------------------------------------------------------
// Streaming pass: one block = (batch b, L-split). 8 waves; each lane owns 8
// interleaved float4 columns of D (d = c*128 + lane*4, c=0..7) and keeps an
// online-softmax state (m, s, acc[32]) in registers across its rows.
// ---------------------------------------------------------------------------
__global__ __launch_bounds__(256)
void attn_partial(const float* __restrict__ enc,
                  const unsigned char* __restrict__ mask,   // jax bool: 1 byte
                  const float* __restrict__ qk,
                  const float* __restrict__ cb,
                  float* __restrict__ part, int L, int Dn) {
  const int b     = blockIdx.y;
  const int split = blockIdx.x;
  const int lane  = threadIdx.x & 31;
  const int wave  = threadIdx.x >> 5;            // 0..7
  const int rowsPerBlock = L / SPLIT;            // 64 for L=2048
  const int rowsPerWave  = rowsPerBlock / 8;     // 8
  const int l0 = split * rowsPerBlock + wave * rowsPerWave;
  const float sc  = rsqrtf((float)Dn);
  const float cbv = cb[b];

  // qk fragment for this lane's 32 columns
  const float4* qk4 = (const float4*)(qk + (size_t)b * Dn);
  float4 qf[8];
#pragma unroll
  for (int c = 0; c < 8; ++c) qf[c] = qk4[c * 32 + lane];

  float4 acc[8];
#pragma unroll
  for (int c = 0; c < 8; ++c) acc[c] = make_float4(0.f, 0.f, 0.f, 0.f);
  float m = -INFINITY, s = 0.0f;

  for (int r = 0; r < rowsPerWave; ++r) {
    const int l = l0 + r;
    const float4* e4 = (const float4*)(enc + ((size_t)b * L + l) * Dn);
    float4 v[8];
#pragma unroll
    for (int c = 0; c < 8; ++c) v[c] = ntload4(e4 + c * 32 + lane);

    float dot = 0.0f;
#pragma unroll
    for (int c = 0; c < 8; ++c) {
      dot = fmaf(qf[c].x, v[c].x, dot);
      dot = fmaf(qf[c].y, v[c].y, dot);
      dot = fmaf(qf[c].z, v[c].z, dot);
      dot = fmaf(qf[c].w, v[c].w, dot);
    }
#pragma unroll
    for (int off = 16; off > 0; off >>= 1) dot += __shfl_xor(dot, off, 32);

    float e = (dot + cbv) * sc;
    if (mask[(size_t)b * L + l]) e = -1.0e10f;   // jnp.where(mask, MASK, e)

    const float mn    = fmaxf(m, e);
    const float scale = __expf(m - mn);          // exp(-inf)=0 on first row
    const float w     = __expf(e - mn);
    s = s * scale + w;
#pragma unroll
    for (int c = 0; c < 8; ++c) {
      acc[c].x = fmaf(w, v[c].x, acc[c].x * scale);
      acc[c].y = fmaf(w, v[c].y, acc[c].y * scale);
      acc[c].z = fmaf(w, v[c].z, acc[c].z * scale);
      acc[c].w = fmaf(w, v[c].w, acc[c].w * scale);
    }
    m = mn;
  }

  // Merge 8 waves via LDS (8 * 1024 * 4B = 32 KB << 320 KB/WGP)
  __shared__ float mArr[8], sArr[8];
  __shared__ float accL[8][1024];
  if (lane == 0) { mArr[wave] = m; sArr[wave] = s; }
#pragma unroll
  for (int c = 0; c < 8; ++c) ((float4*)accL[wave])[c * 32 + lane] = acc[c];
  __syncthreads();

  float mb = -INFINITY;
#pragma unroll
  for (int w2 = 0; w2 < 8; ++w2) mb = fmaxf(mb, mArr[w2]);
  float wsc[8];
  float sb = 0.0f;
#pragma unroll
  for (int w2 = 0; w2 < 8; ++w2) {
    wsc[w2] = __expf(mArr[w2] - mb);
    sb = fmaf(sArr[w2], wsc[w2], sb);
  }

  // thread t reduces columns d = 4t..4t+3 across the 8 waves
  const int t = threadIdx.x;
  float4 outv = make_float4(0.f, 0.f, 0.f, 0.f);
#pragma unroll
  for (int w2 = 0; w2 < 8; ++w2) {
    float4 a = ((float4*)accL[w2])[t];
    outv.x = fmaf(wsc[w2], a.x, outv.x);
    outv.y = fmaf(wsc[w2], a.y, outv.y);
    outv.z = fmaf(wsc[w2], a.z, outv.z);
    outv.w = fmaf(wsc[w2], a.w, outv.w);
  }
  float* dst = part + ((size_t)b * SPLIT + split) * PART_STRIDE;
  ((float4*)dst)[t] = outv;
  if (t == 0) { dst[1024] = mb; dst[1025] = sb; }
}

// Combine SPLIT partials per batch row: out[b,:] = sum(acc_i*e^(m_i-m)) / s
__global__ __launch_bounds__(256)
void attn_combine(const float* __restrict__ part, float* __restrict__ out, int Dn) {
  const int b = blockIdx.x;
  const int t = threadIdx.x;
  const float* base = part + (size_t)b * SPLIT * PART_STRIDE;

  float mg = -INFINITY;
#pragma unroll
  for (int sp = 0; sp < SPLIT; ++sp)
    mg = fmaxf(mg, base[sp * PART_STRIDE + 1024]);

  float sg = 0.0f;
  float4 acc = make_float4(0.f, 0.f, 0.f, 0.f);
#pragma unroll 4
  for (int sp = 0; sp < SPLIT; ++sp) {
    const float* p = base + sp * PART_STRIDE;
    const float w = __expf(p[1024] - mg);
    sg = fmaf(p[1025], w, sg);
    float4 a = ((const float4*)p)[t];
    acc.x = fmaf(w, a.x, acc.x);
    acc.y = fmaf(w, a.y, acc.y);
    acc.z = fmaf(w, a.z, acc.z);
    acc.w = fmaf(w, a.w, acc.w);
  }
  const float inv = 1.0f / sg;   // row 0 is always unmasked -> sg > 0
  float4 r = make_float4(acc.x * inv, acc.y * inv, acc.z * inv, acc.w * inv);
  ((float4*)(out + (size_t)b * Dn))[t] = r;
}

// ---------------------------------------------------------------------------
// Inputs (setup_inputs order):
//  0 encoder_output [B,L,D] f32   1 decoder_hidden [B,D] f32
//  2 mask [B,L] bool(1B)          3 max_src_length (scalar)
//  4 Wq [D,D] f32                 5 bq [D] f32
//  6 Wk [D,D] f32                 7 bk [D] f32
// Output: context [B,1,D] f32
// ---------------------------------------------------------------------------
extern "C" void kernel_launch(void* const* d_in, const int* in_sizes, int n_in,
                              void* d_out, int out_size, void* d_ws, size_t ws_size,
                              hipStream_t stream) {
  const float*         enc  = (const float*)d_in[0];
  const float*         h    = (const float*)d_in[1];
  const unsigned char* mask = (const unsigned char*)d_in[2];
  const float*         Wq   = (const float*)d_in[4];
  const float*         bq   = (const float*)d_in[5];
  const float*         Wk   = (const float*)d_in[6];
  const float*         bk   = (const float*)d_in[7];
  float*               out  = (float*)d_out;

  const int D = in_sizes[0] / in_sizes[2];   // (B*L*D)/(B*L)
  const int B = in_sizes[1] / D;             // (B*D)/D
  const int L = in_sizes[2] / B;             // (B*L)/B

  // Workspace layout (floats): q[B*D] | qk[B*D] | cb[pad 64] | partials
  float* ws = (float*)d_ws;
  float* q_ws   = ws;
  float* qk_ws  = q_ws + (size_t)B * D;
  float* cb_ws  = qk_ws + (size_t)B * D;
  float* part   = cb_ws + 64;                // 16B-aligned

  // 1) q = h @ Wq + bq            (WMMA f32, M=B N=D K=D, B stored [K,N])
  attn_wmma_gemm_f32<0><<<dim3(D / 16, B / 16), 32, 0, stream>>>(
      h, Wq, bq, q_ws, B, D, D);
  // 2) qk = q @ Wk^T              (WMMA f32, Wk stored [D,D] row-major)
  attn_wmma_gemm_f32<1><<<dim3(D / 16, B / 16), 32, 0, stream>>>(
      q_ws, Wk, nullptr, qk_ws, B, D, D);
  // 3) cb[b] = q[b,:] . bk
  attn_row_dot<<<B, 256, 0, stream>>>(q_ws, bk, cb_ws, D);
  // 4) single streaming pass over encoder_output with online softmax
  attn_partial<<<dim3(SPLIT, B), 256, 0, stream>>>(
      enc, mask, qk_ws, cb_ws, part, L, D);
  // 5) combine split partials -> context
  attn_combine<<<B, 256, 0, stream>>>(part, out, D);
}